// SVDHead_47931835023385
// MI455X (gfx1250) — compile-verified
//
#include <hip/hip_runtime.h>
#include <stdint.h>

// Batched weighted Kabsch / Procrustes for MI455X (gfx1250, wave32).
// One batch per lane. Tiles staged global->LDS with CDNA5 async copies
// (GLOBAL_LOAD_ASYNC_TO_LDS_B128 + s_wait_asynccnt) for full coalescing.
// Rotation extracted via Horn's quaternion method (top eigenvector of the
// 4x4 N matrix by 6 cyclic Jacobi sweeps) == V*diag(1,1,det)*U^T from SVD.

#define TB 64              // threads per block == batches per tile
#define NP 20              // points per batch (reference setup)
#define AF (TB * NP * 3)   // floats per A/B tile = 3840 (15360 B)
#define WF (TB * NP)       // floats per weight tile = 1280 (5120 B)
#define ACH (AF / 4)       // 16B chunks per A/B tile = 960 -> 15/thread
#define WCH (WF / 4)       // 16B chunks per weight tile = 320 -> 5/thread

#define LDS_AS __attribute__((address_space(3)))
#define GLB_AS __attribute__((address_space(1)))

typedef __attribute__((__vector_size__(4 * sizeof(int)))) int v4i;

static __device__ __forceinline__ void async_copy16(const float* g, float* l) {
#if __has_builtin(__builtin_amdgcn_global_load_async_to_lds_b128)
  __builtin_amdgcn_global_load_async_to_lds_b128(
      (GLB_AS v4i*)(void*)g, (LDS_AS v4i*)(void*)l, 0, 0);
#else
  asm volatile("global_load_async_to_lds_b128 %0, %1, off"
               :
               : "v"((uint32_t)(uintptr_t)(LDS_AS void*)(void*)l), "v"(g)
               : "memory");
#endif
}

static __device__ __forceinline__ void wait_async0() {
#if __has_builtin(__builtin_amdgcn_s_wait_asynccnt)
  __builtin_amdgcn_s_wait_asynccnt(0);
#else
  asm volatile("s_wait_asynccnt 0" ::: "memory");
#endif
}

// One cyclic-Jacobi rotation on symmetric 4x4 Nm, accumulating eigenvectors
// into Qm columns. P < R are template constants so Nm/Qm stay in registers.
template <int P, int R>
static __device__ __forceinline__ void jrot(float Nm[4][4], float Qm[4][4]) {
  float apq = Nm[P][R];
  float app = Nm[P][P], aqq = Nm[R][R];
  float th  = 0.5f * (aqq - app);
  float sg  = (th >= 0.0f) ? 1.0f : -1.0f;
  float den = fabsf(th) + sqrtf(th * th + apq * apq);
  float t   = (den > 1e-37f) ? (sg * apq / den) : 0.0f;   // tan of rotation
  float c   = rsqrtf(1.0f + t * t);
  float s   = t * c;
  Nm[P][P] = app - t * apq;
  Nm[R][R] = aqq + t * apq;
  Nm[P][R] = 0.0f; Nm[R][P] = 0.0f;
#pragma unroll
  for (int k = 0; k < 4; ++k) {
    if (k == P || k == R) continue;
    float akp = Nm[k][P], akq = Nm[k][R];
    float np = c * akp - s * akq;
    float nq = s * akp + c * akq;
    Nm[k][P] = np; Nm[P][k] = np;
    Nm[k][R] = nq; Nm[R][k] = nq;
  }
#pragma unroll
  for (int k = 0; k < 4; ++k) {
    float qkp = Qm[k][P], qkq = Qm[k][R];
    Qm[k][P] = c * qkp - s * qkq;
    Qm[k][R] = s * qkp + c * qkq;
  }
}

__global__ __launch_bounds__(TB) void kabsch64(
    const float* __restrict__ A, const float* __restrict__ B,
    const float* __restrict__ Wt, float* __restrict__ out, int bs) {
  __shared__ float sA[AF];
  __shared__ float sB[AF];
  __shared__ float sW[WF];

  const int tid = threadIdx.x;
  const int b0  = blockIdx.x * TB;
  const int nb  = (bs - b0 < TB) ? (bs - b0) : TB;

  const float* gA = A  + (size_t)b0 * (NP * 3);
  const float* gB = B  + (size_t)b0 * (NP * 3);
  const float* gW = Wt + (size_t)b0 * NP;

  const int maxA = nb * NP * 3 * 4 - 16;  // clamp for partial tail tile
  const int maxW = nb * NP * 4 - 16;

#pragma unroll
  for (int k = 0; k < ACH / TB; ++k) {
    int off  = (k * TB + tid) * 16;       // coalesced: 512B per wave per op
    int goff = off > maxA ? maxA : off;
    async_copy16((const float*)((const char*)gA + goff),
                 (float*)((char*)sA + off));
    async_copy16((const float*)((const char*)gB + goff),
                 (float*)((char*)sB + off));
  }
#pragma unroll
  for (int k = 0; k < WCH / TB; ++k) {
    int off  = (k * TB + tid) * 16;
    int goff = off > maxW ? maxW : off;
    async_copy16((const float*)((const char*)gW + goff),
                 (float*)((char*)sW + off));
  }
  wait_async0();
  __syncthreads();

  const int b = b0 + tid;
  if (b >= bs) return;

  const float* pa = sA + tid * (NP * 3);
  const float* pb = sB + tid * (NP * 3);
  const float* pw = sW + tid * NP;

  // Single pass: wsum, weighted sums, raw second moments S = sum w*a*b^T.
  float S00=0,S01=0,S02=0,S10=0,S11=0,S12=0,S20=0,S21=0,S22=0;
  float sax=0,say=0,saz=0, sbx=0,sby=0,sbz=0, wsum=0;
#pragma unroll
  for (int j = 0; j < NP; ++j) {
    float w = pw[j]; w = (w < 0.0f) ? 0.0f : w;   // WEIGHT_THRESHOLD = 0
    float ax = pa[3*j+0], ay = pa[3*j+1], az = pa[3*j+2];
    float bx = pb[3*j+0], by = pb[3*j+1], bz = pb[3*j+2];
    wsum += w;
    float wax = w*ax, way = w*ay, waz = w*az;
    sax += wax; say += way; saz += waz;
    sbx += w*bx; sby += w*by; sbz += w*bz;
    S00 += wax*bx; S01 += wax*by; S02 += wax*bz;
    S10 += way*bx; S11 += way*by; S12 += way*bz;
    S20 += waz*bx; S21 += waz*by; S22 += waz*bz;
  }
  // H = S - coef * sA sB^T with centroids cX = sX/(wsum+EPS); coef folds the
  // centering exactly as the reference (EPS = 1e-6 in the denominator).
  float Wp   = wsum + 1e-6f;
  float inv  = 1.0f / Wp;
  float coef = (wsum + 2e-6f) * inv * inv;
  float H00 = S00 - coef*sax*sbx, H01 = S01 - coef*sax*sby, H02 = S02 - coef*sax*sbz;
  float H10 = S10 - coef*say*sbx, H11 = S11 - coef*say*sby, H12 = S12 - coef*say*sbz;
  float H20 = S20 - coef*saz*sbx, H21 = S21 - coef*saz*sby, H22 = S22 - coef*saz*sbz;
  float cax = sax*inv, cay = say*inv, caz = saz*inv;
  float cbx = sbx*inv, cby = sby*inv, cbz = sbz*inv;

  // Horn's 4x4 symmetric key matrix; its top eigenvector is the quaternion of
  // argmax_{R in SO(3)} tr(R H) == V diag(1,1,det) U^T from the SVD of H.
  float Nm[4][4];
  Nm[0][0] = H00 + H11 + H22;
  Nm[0][1] = Nm[1][0] = H12 - H21;
  Nm[0][2] = Nm[2][0] = H20 - H02;
  Nm[0][3] = Nm[3][0] = H01 - H10;
  Nm[1][1] = H00 - H11 - H22;
  Nm[1][2] = Nm[2][1] = H01 + H10;
  Nm[1][3] = Nm[3][1] = H20 + H02;
  Nm[2][2] = -H00 + H11 - H22;
  Nm[2][3] = Nm[3][2] = H12 + H21;
  Nm[3][3] = -H00 - H11 + H22;

  float Qm[4][4] = {{1,0,0,0},{0,1,0,0},{0,0,1,0},{0,0,0,1}};
  for (int sw = 0; sw < 6; ++sw) {           // quadratic convergence
    jrot<0,1>(Nm, Qm); jrot<0,2>(Nm, Qm); jrot<0,3>(Nm, Qm);
    jrot<1,2>(Nm, Qm); jrot<1,3>(Nm, Qm); jrot<2,3>(Nm, Qm);
  }

  // Select eigenvector of the largest eigenvalue (branchless cndmask chain).
  float bl = Nm[0][0];
  float qw = Qm[0][0], qx = Qm[1][0], qy = Qm[2][0], qz = Qm[3][0];
#pragma unroll
  for (int i = 1; i < 4; ++i) {
    bool better = Nm[i][i] > bl;
    bl = better ? Nm[i][i] : bl;
    qw = better ? Qm[0][i] : qw;
    qx = better ? Qm[1][i] : qx;
    qy = better ? Qm[2][i] : qy;
    qz = better ? Qm[3][i] : qz;
  }
  float qn = rsqrtf(qw*qw + qx*qx + qy*qy + qz*qz + 1e-30f);
  qw *= qn; qx *= qn; qy *= qn; qz *= qn;

  float R00 = qw*qw + qx*qx - qy*qy - qz*qz;
  float R01 = 2.0f*(qx*qy - qw*qz);
  float R02 = 2.0f*(qx*qz + qw*qy);
  float R10 = 2.0f*(qx*qy + qw*qz);
  float R11 = qw*qw - qx*qx + qy*qy - qz*qz;
  float R12 = 2.0f*(qy*qz - qw*qx);
  float R20 = 2.0f*(qx*qz - qw*qy);
  float R21 = 2.0f*(qy*qz + qw*qx);
  float R22 = qw*qw - qx*qx - qy*qy + qz*qz;

  float tx = cbx - (R00*cax + R01*cay + R02*caz);
  float ty = cby - (R10*cax + R11*cay + R12*caz);
  float tz = cbz - (R20*cax + R21*cay + R22*caz);

  float4* o = (float4*)(out + (size_t)b * 16);
  o[0] = make_float4(R00, R01, R02, tx);
  o[1] = make_float4(R10, R11, R12, ty);
  o[2] = make_float4(R20, R21, R22, tz);
  o[3] = make_float4(0.0f, 0.0f, 0.0f, 1.0f);
}

extern "C" void kernel_launch(void* const* d_in, const int* in_sizes, int n_in,
                              void* d_out, int out_size, void* d_ws, size_t ws_size,
                              hipStream_t stream) {
  (void)in_sizes; (void)n_in; (void)d_ws; (void)ws_size;
  const float* A = (const float*)d_in[0];
  const float* B = (const float*)d_in[1];
  const float* W = (const float*)d_in[2];
  float* out = (float*)d_out;
  const int bs = out_size / 16;                 // [bs,4,4] output
  const int grid = (bs + TB - 1) / TB;
  kabsch64<<<grid, TB, 0, stream>>>(A, B, W, out, bs);
}